// DeepHitLoss_46832323396161
// MI455X (gfx1250) — compile-verified
//
#include <hip/hip_runtime.h>
#include <hip/hip_bf16.h>

// DeepHit loss for MI455X (gfx1250, wave32).
//   K0: zero counters.
//   K1 (prep): per-sample cumsum over T=64; LL atomics; compact censored list
//       (t_i, d_i=cum[i,t_i] per risk) and event list (t_j + full cum row per
//       risk) into workspace.
//   K2 (pairwise): persistent wave-tasked kernel. Task = (risk, censored 16-row
//       strip, chunk of 8 event 16-col tiles). Per strip: hoist one-hot A
//       fragments (exact 0/1 f32) + epilogue t_i/d_i. Per tile: G[m][n] =
//       cum[j_n][t_i(m)] via 16x v_wmma_f32_16x16x4_f32 in TWO independent
//       accumulator chains (even/odd K) so the XDL pipe overlaps instead of
//       stalling on the D->C dependency. B fragments use HALF-TILE double
//       buffering (2 x 8 fragments = 32 VGPRs instead of 64) to stay under the
//       ~128-VGPR budget -- the full-tile ping/pong of the previous round
//       spilled B fragments to scratch inside the WMMA chain. Per-tile t_n is
//       prefetched with the B halves so the epilogue is pure-register.
//       EXEC all-ones throughout (WMMA requirement); ragged edges VALU-masked.
//   K3 (finalize): scalar combine.
//
// ws layout (~4.33 MB):
//   [0] int nc  [4] int ne  [8] int llcnt[2]  [16] float llsum[2]  [24] float rlsum
//   [64]     int   tc[8192]
//   [32832]  int   te[8192]
//   [65600]  float dcv[2][8192]
//   [131136] float ecum[2][8192][64]

#define B_N 8192
#define T_N 64
#define EJ_CHUNK 8

typedef __attribute__((ext_vector_type(2))) float v2f;
typedef __attribute__((ext_vector_type(8))) float v8f;

__global__ void dh_init(int* nc, int* ne, int* llcnt, float* llsum, float* rlsum) {
    if (threadIdx.x == 0) {
        *nc = 0; *ne = 0;
        llcnt[0] = 0; llcnt[1] = 0;
        llsum[0] = 0.f; llsum[1] = 0.f;
        *rlsum = 0.f;
    }
}

__global__ void dh_prep(const float* __restrict__ pred,   // (2,B,T)
                        const int*   __restrict__ tbin,   // (B,)
                        const int*   __restrict__ rind,   // (B,2)
                        const int*   __restrict__ ev,     // (B,)
                        int* nc, int* ne, int* llcnt, float* llsum,
                        int* __restrict__ tc, int* __restrict__ te,
                        float* __restrict__ dcv, float* __restrict__ ecum) {
    int i = blockIdx.x * blockDim.x + threadIdx.x;
    if (i >= B_N) return;
    int  t = tbin[i];
    bool e = (ev[i] != 0);

    int slot_e = -1, slot_c = -1;
    if (e) slot_e = atomicAdd(ne, 1);
    else   slot_c = atomicAdd(nc, 1);
    if (e) te[slot_e] = t; else tc[slot_c] = t;

    #pragma unroll
    for (int r = 0; r < 2; ++r) {
        const float* row = pred + ((size_t)r * B_N + (size_t)i) * T_N;
        float run = 0.f, pt = 0.f, dt = 0.f;
        float* erow = (slot_e >= 0)
                    ? (ecum + ((size_t)r * B_N + (size_t)slot_e) * T_N) : nullptr;
        for (int k = 0; k < T_N; ++k) {
            float v = row[k];
            run += v;
            if (k == t) { pt = v; dt = run; }
            if (erow) erow[k] = run;
        }
        if (slot_c >= 0) dcv[r * B_N + slot_c] = dt;
        if (e && (rind[i * 2 + r] != 0)) {
            atomicAdd(&llsum[r], logf(pt + 1e-8f));
            atomicAdd(&llcnt[r], 1);
        }
    }
}

// Load clause: 8 B fragments of half `h_` (K rows 32*h_..32*h_+31) of tile ej_.
#define LOAD_HALF(ej_, h_, B_)                                                 \
    do {                                                                       \
        const float* brow_ = ecum +                                            \
            ((size_t)r * B_N + (size_t)((ej_) * 16 + mrow)) * T_N              \
            + (h_) * 32 + kkbase;                                              \
        _Pragma("unroll")                                                      \
        for (int k_ = 0; k_ < 8; ++k_)                                         \
            (B_)[k_] = *(const v2f*)(brow_ + 4 * k_);                          \
    } while (0)

// 8 WMMAs of half `h_` feeding the two independent accumulator chains.
#define WMMA_HALF(h_, B_)                                                      \
    do {                                                                       \
        _Pragma("unroll")                                                      \
        for (int k_ = 0; k_ < 4; ++k_) {                                       \
            accE = __builtin_amdgcn_wmma_f32_16x16x4_f32(false,                \
                afr[(h_) * 8 + 2 * k_], false, (B_)[2 * k_],                   \
                (short)0, accE, false, false);                                 \
            accO = __builtin_amdgcn_wmma_f32_16x16x4_f32(false,                \
                afr[(h_) * 8 + 2 * k_ + 1], false, (B_)[2 * k_ + 1],           \
                (short)0, accO, false, false);                                 \
        }                                                                      \
    } while (0)

__global__ void dh_pairwise(const int* __restrict__ nc_p, const int* __restrict__ ne_p,
                            const int* __restrict__ tc, const int* __restrict__ te,
                            const float* __restrict__ dcv, const float* __restrict__ ecum,
                            float* __restrict__ rlsum) {
    const int nc = *nc_p;
    const int ne = *ne_p;
    if (nc == 0 || ne == 0) return;  // uniform

    const int lane = threadIdx.x & 31;
    const int wib  = threadIdx.x >> 5;
    const int gwave  = blockIdx.x * 8 + wib;
    const int nwaves = gridDim.x * 8;

    const int nct = (nc + 15) >> 4;                 // censored 16-row strips
    const int net = (ne + 15) >> 4;                 // event 16-col tiles
    const int nch = (net + EJ_CHUNK - 1) / EJ_CHUNK;
    const int ntasks = 2 * nct * nch;               // (risk, strip, chunk)

    const int mrow   = lane & 15;                   // A row m / B col n for loads
    const int kkbase = (lane < 16) ? 0 : 2;         // K sub-rows held by this half-wave
    const int n      = lane & 15;                   // epilogue column
    const int mbase  = (lane < 16) ? 0 : 8;         // epilogue row base

    float lsum = 0.f;

    for (int task = gwave; task < ntasks; task += nwaves) {
        int strip = task / nch;                     // (risk, ci)
        int chunk = task - strip * nch;
        int r  = strip / nct;
        int ci = strip - r * nct;

        // ---- hoisted per-strip state ------------------------------------
        int t_m = tc[ci * 16 + mrow];
        v2f afr[16];                                // one-hot A, exact 0/1 f32
        #pragma unroll
        for (int k = 0; k < 16; ++k) {
            int k0 = 4 * k + kkbase;
            afr[k].x = (t_m == k0)     ? 1.f : 0.f;
            afr[k].y = (t_m == k0 + 1) ? 1.f : 0.f;
        }
        int   t_i8[8];                              // validity folded: -1 = never ok
        float d_i8[8];
        #pragma unroll
        for (int v = 0; v < 8; ++v) {
            int gi = ci * 16 + mbase + v;
            bool vm = (gi < nc);
            t_i8[v] = vm ? tc[gi] : -1;
            d_i8[v] = dcv[r * B_N + gi];
        }

        // ---- sweep event tiles: half-tile pipelined B + t_n -------------
        int ej_lo = chunk * EJ_CHUNK;
        int ej_hi = min(net, ej_lo + EJ_CHUNK);

        v2f bA[8], bB[8];
        int ej = ej_lo;
        int tn_cur, tn_nxt;
        LOAD_HALF(ej, 0, bA);                       // prologue: half-0 of first tile
        tn_cur = te[ej * 16 + n];
        for (;;) {
            LOAD_HALF(ej, 1, bB);                   // clause: half-1 of this tile

            v8f accE = {0.f, 0.f, 0.f, 0.f, 0.f, 0.f, 0.f, 0.f};
            v8f accO = {0.f, 0.f, 0.f, 0.f, 0.f, 0.f, 0.f, 0.f};
            WMMA_HALF(0, bA);                       // consumes bA

            bool more = (ej + 1 < ej_hi);
            if (more) {                             // prefetch next tile's half-0
                LOAD_HALF(ej + 1, 0, bA);
                tn_nxt = te[(ej + 1) * 16 + n];
            }
            WMMA_HALF(1, bB);                       // consumes bB

            // pure-register masked sigmoid epilogue
            bool vn = ((ej * 16 + n) < ne);
            #pragma unroll
            for (int v = 0; v < 8; ++v) {
                bool  ok = vn && (tn_cur <= t_i8[v]);
                float x  = (d_i8[v] - accE[v] - accO[v]) * 10.0f;  // diff / sigma
                float s  = __builtin_amdgcn_rcpf(1.0f + __expf(-x));
                lsum += ok ? s : 0.0f;
            }

            if (!more) break;
            tn_cur = tn_nxt;
            ++ej;
        }
    }

    // wave32 shuffle reduction, then block reduction, one atomic per block
    #pragma unroll
    for (int off = 16; off > 0; off >>= 1)
        lsum += __shfl_down(lsum, off, 32);

    __shared__ float red[8];
    if (lane == 0) red[wib] = lsum;
    __syncthreads();
    if (threadIdx.x == 0) {
        float s = 0.f;
        #pragma unroll
        for (int w = 0; w < 8; ++w) s += red[w];
        atomicAdd(rlsum, s);
    }
}

__global__ void dh_fin(const float* llsum, const int* llcnt, const float* rlsum,
                       float* out) {
    if (threadIdx.x == 0) {
        float ll = 0.f;
        #pragma unroll
        for (int r = 0; r < 2; ++r)
            ll += (llcnt[r] > 0) ? (-llsum[r] / (float)llcnt[r]) : 0.f;
        out[0] = 0.5f * ll + 0.5f * rlsum[0];
    }
}

extern "C" void kernel_launch(void* const* d_in, const int* in_sizes, int n_in,
                              void* d_out, int out_size, void* d_ws, size_t ws_size,
                              hipStream_t stream) {
    const float* pred = (const float*)d_in[0];   // risk_outputs (2,8192,64) f32
    const int*   tbin = (const int*)d_in[1];     // time_bins (8192,)
    const int*   rind = (const int*)d_in[2];     // risk_indicators (8192,2)
    const int*   ev   = (const int*)d_in[3];     // event_indicators (8192,)

    char* ws = (char*)d_ws;
    int*   nc    = (int*)(ws + 0);
    int*   ne    = (int*)(ws + 4);
    int*   llcnt = (int*)(ws + 8);
    float* llsum = (float*)(ws + 16);
    float* rlsum = (float*)(ws + 24);
    int*   tc    = (int*)(ws + 64);
    int*   te    = (int*)(ws + 32832);
    float* dcv   = (float*)(ws + 65600);
    float* ecum  = (float*)(ws + 131136);

    dh_init<<<1, 32, 0, stream>>>(nc, ne, llcnt, llsum, rlsum);
    dh_prep<<<B_N / 256, 256, 0, stream>>>(pred, tbin, rind, ev,
                                           nc, ne, llcnt, llsum, tc, te, dcv, ecum);
    dh_pairwise<<<2048, 256, 0, stream>>>(nc, ne, tc, te, dcv, ecum, rlsum);
    dh_fin<<<1, 32, 0, stream>>>(llsum, llcnt, rlsum, (float*)d_out);
}